// NeuromodulatedAttention_18528488915402
// MI455X (gfx1250) — compile-verified
//
#include <hip/hip_runtime.h>
#include <hip/hip_bf16.h>
#include <math.h>
#include <stdint.h>

// Problem constants (reference: B=2, T=2048, C=1024, H=16, D=64)
#define NB 2
#define TT 2048
#define NH 16
#define DH 64
#define HD 1024   // NH*DH, also C

typedef float v2f __attribute__((ext_vector_type(2)));
typedef float v4f __attribute__((ext_vector_type(4)));
typedef float v8f __attribute__((ext_vector_type(8)));

__device__ __forceinline__ v8f wmma_f32(v2f a, v2f b, v8f c) {
  // V_WMMA_F32_16X16X4_F32 : D = A(16x4) * B(4x16) + C(16x16), fp32
  return __builtin_amdgcn_wmma_f32_16x16x4_f32(false, a, false, b, (short)0, c,
                                               false, false);
}

// Async global->LDS copy, 16B per lane (GLOBAL_LOAD_ASYNC_TO_LDS_B128).
// LDS dest address = low 32 bits of the generic pointer (flat LDS aperture).
__device__ __forceinline__ void async_copy_b128(void* lds_dst,
                                                const void* gsrc) {
  uint32_t l = (uint32_t)(uintptr_t)lds_dst;
  uint64_t g = (uint64_t)(uintptr_t)gsrc;
  asm volatile("global_load_async_to_lds_b128 %0, %1, off" ::"v"(l), "v"(g)
               : "memory");
}
__device__ __forceinline__ void wait_async() {
  asm volatile("s_wait_asynccnt 0x0" ::: "memory");
}

// reductions across the 16-lane half-groups (wave32): xor masks 1,2,4,8 stay
// inside each 16-lane half, matching the N-striping of the C/D fragment.
__device__ __forceinline__ float hmax16(float v) {
  v = fmaxf(v, __shfl_xor(v, 1, 32));
  v = fmaxf(v, __shfl_xor(v, 2, 32));
  v = fmaxf(v, __shfl_xor(v, 4, 32));
  v = fmaxf(v, __shfl_xor(v, 8, 32));
  return v;
}
__device__ __forceinline__ float hsum16(float v) {
  v += __shfl_xor(v, 1, 32);
  v += __shfl_xor(v, 2, 32);
  v += __shfl_xor(v, 4, 32);
  v += __shfl_xor(v, 8, 32);
  return v;
}

// ---------------------------------------------------------------------------
// Generic fp32 GEMM + bias:  C[M,N] = A[M,K] @ B[K,N] + bias[N]
// 64x64 block tile, 256 threads = 8 waves, each wave two 16x16 WMMA tiles.
// Double-buffered LDS staging via async global->LDS DMA (ASYNCcnt).
// ---------------------------------------------------------------------------
__global__ __launch_bounds__(256) void gemm_bias_kernel(
    const float* __restrict__ A, const float* __restrict__ Bm,
    const float* __restrict__ bias, float* __restrict__ Cm, int M, int N,
    int K) {
  __shared__ float As[2][64][20];  // 80B rows: 16B aligned, banks distinct
  __shared__ float Bs[2][16][68];  // 272B rows: 16B aligned
  const int tid = threadIdx.x;
  const int wave = tid >> 5, lane = tid & 31;
  const int wr = wave >> 1, wc = wave & 1;  // 4x2 wave grid over 64x64
  const int m0 = blockIdx.y * 64, n0 = blockIdx.x * 64;
  const int mrow = lane & 15;
  const int kk = (lane >> 4) * 2;  // K-pair per lane-half

  const int ar = tid >> 2, ac = (tid & 3) * 4;   // A stage: 64 x (4x4)
  const int br = tid >> 4, bc = (tid & 15) * 4;  // B stage: 16 x (16x4)

  // prologue: kick off tile 0
  async_copy_b128(&As[0][ar][ac], &A[(size_t)(m0 + ar) * K + ac]);
  async_copy_b128(&Bs[0][br][bc], &Bm[(size_t)br * N + n0 + bc]);

  v8f acc0 = {}, acc1 = {};
  for (int k0 = 0; k0 < K; k0 += 16) {
    const int cur = (k0 >> 4) & 1;
    wait_async();
    __syncthreads();
    if (k0 + 16 < K) {  // issue next tile's DMA; overlaps with WMMAs below
      async_copy_b128(&As[cur ^ 1][ar][ac],
                      &A[(size_t)(m0 + ar) * K + k0 + 16 + ac]);
      async_copy_b128(&Bs[cur ^ 1][br][bc],
                      &Bm[(size_t)(k0 + 16 + br) * N + n0 + bc]);
    }
#pragma unroll
    for (int ks = 0; ks < 16; ks += 4) {
      v2f a = *(const v2f*)&As[cur][wr * 16 + mrow][ks + kk];
      v2f b0, b1;
      b0.x = Bs[cur][ks + kk][wc * 32 + mrow];
      b0.y = Bs[cur][ks + kk + 1][wc * 32 + mrow];
      b1.x = Bs[cur][ks + kk][wc * 32 + 16 + mrow];
      b1.y = Bs[cur][ks + kk + 1][wc * 32 + 16 + mrow];
      acc0 = wmma_f32(a, b0, acc0);
      acc1 = wmma_f32(a, b1, acc1);
    }
    __syncthreads();
  }
#pragma unroll
  for (int i = 0; i < 8; ++i) {
    int m = m0 + wr * 16 + i + ((lane < 16) ? 0 : 8);
    int nc = n0 + wc * 32 + mrow;
    Cm[(size_t)m * N + nc] = acc0[i] + bias[nc];
    Cm[(size_t)m * N + nc + 16] = acc1[i] + bias[nc + 16];
  }
}

// ---------------------------------------------------------------------------
// Modulator: temp_scale[b,h] = 1 + softplus(mod0), diag[b,h] = mod1
// ---------------------------------------------------------------------------
__global__ void modulator_kernel(const float* __restrict__ p,
                                 const float* __restrict__ wmod,
                                 const float* __restrict__ bmod,
                                 float* __restrict__ temp_out,
                                 float* __restrict__ ws_rtemp,
                                 float* __restrict__ ws_diag) {
  int idx = threadIdx.x;
  if (idx >= NB * NH) return;
  int b = idx / NH, h = idx % NH;
  float ps = p[b];
  float m0 = ps * wmod[2 * h] + bmod[2 * h];
  float m1 = ps * wmod[2 * h + 1] + bmod[2 * h + 1];
  float sp = fmaxf(m0, 0.f) + log1pf(__expf(-fabsf(m0)));  // stable softplus
  float temp = 1.f + sp;
  temp_out[idx] = temp;          // goes straight to d_out tail
  ws_rtemp[idx] = 1.f / temp;
  ws_diag[idx] = m1;
}

// ---------------------------------------------------------------------------
// Pass 1: per-(b,h) row stats over t:  m[s]=max_t score, l[s]=sum_t exp(.-m)
// score[s,t] = (q_t . k_s) * 0.125 / temp + diag*(s==t), masked over t.
// Block: 128 s-rows (8 waves x 16), sweep t in 16-wide WMMA tiles.
// Q tiles double-buffered via async DMA; softmax tail overlaps the copy.
// ---------------------------------------------------------------------------
__global__ __launch_bounds__(256) void attn_stats_kernel(
    const float* __restrict__ q, const float* __restrict__ k,
    const unsigned char* __restrict__ mask, const float* __restrict__ rtemp,
    const float* __restrict__ diagb, float* __restrict__ mstat,
    float* __restrict__ lstat) {
  __shared__ float Qs[2][16][68];
  const int tid = threadIdx.x;
  const int wave = tid >> 5, lane = tid & 31;
  const int bh = blockIdx.y;
  const int b = bh / NH;
  const int h = bh % NH;
  const int sbase = blockIdx.x * 128 + wave * 16;
  const int mrow = lane & 15;
  const int kk = (lane >> 4) * 2;
  const size_t base = ((size_t)b * TT * NH + h) * DH;  // + t*HD + d
  const float scl = 0.125f * rtemp[bh];
  const float diag = diagb[bh];
  const int qr = tid >> 4, qc = (tid & 15) * 4;  // Q stage: 16 x (16x4)

  // K-rows for this wave's 16 s-rows: register-resident A fragments (reused
  // across the entire t sweep). 16 d-steps of K=4; 8B global loads.
  v2f ka[16];
  {
    const float* krow = k + base + (size_t)(sbase + mrow) * HD;
#pragma unroll
    for (int ds = 0; ds < 16; ++ds) ka[ds] = *(const v2f*)&krow[ds * 4 + kk];
  }
  float rm[8], rl[8];
#pragma unroll
  for (int i = 0; i < 8; ++i) {
    rm[i] = -3.0e38f;
    rl[i] = 0.f;
  }

  async_copy_b128(&Qs[0][qr][qc], &q[base + (size_t)qr * HD + qc]);
  for (int tt = 0; tt < TT; tt += 16) {
    const int cur = (tt >> 4) & 1;
    wait_async();
    __syncthreads();
    if (tt + 16 < TT)
      async_copy_b128(&Qs[cur ^ 1][qr][qc],
                      &q[base + (size_t)(tt + 16 + qr) * HD + qc]);
    v8f acc = {};
#pragma unroll
    for (int ds = 0; ds < 16; ++ds) {
      v2f bfr = *(const v2f*)&Qs[cur][mrow][ds * 4 + kk];
      acc = wmma_f32(ka[ds], bfr, acc);
    }
    int t = tt + mrow;
    bool mok = mask[b * TT + t] != 0;
#pragma unroll
    for (int i = 0; i < 8; ++i) {
      int s = sbase + i + ((lane < 16) ? 0 : 8);
      float val = acc[i] * scl;
      if (s == t) val += diag;
      if (!mok) val = -1.0e9f;
      float tmax = hmax16(val);
      float nm = fmaxf(rm[i], tmax);
      float ex = __expf(val - nm);
      float ts = hsum16(ex);
      rl[i] = rl[i] * __expf(rm[i] - nm) + ts;
      rm[i] = nm;
    }
    __syncthreads();
  }
  if (mrow == 0) {
#pragma unroll
    for (int i = 0; i < 8; ++i) {
      int s = sbase + i + ((lane < 16) ? 0 : 8);
      mstat[(size_t)bh * TT + s] = rm[i];
      lstat[(size_t)bh * TT + s] = rl[i];
    }
  }
}

// ---------------------------------------------------------------------------
// Pass 2: O[t,d] = sum_s P[s,t] * v[s,d], P = exp(score - m[s]) / l[s].
// Block: 64 t-rows, all 64 d. Loop s in chunks of 32:
//   phase 1: 8 waves compute 32x64 score tile (WMMA over d), normalize,
//            write P^T into LDS.
//   phase 2: 8 waves accumulate O += P^T V (WMMA over s) from LDS.
// V chunks staged via async global->LDS DMA.
// ---------------------------------------------------------------------------
__global__ __launch_bounds__(256) void attn_out_kernel(
    const float* __restrict__ q, const float* __restrict__ k,
    const float* __restrict__ v, const unsigned char* __restrict__ mask,
    const float* __restrict__ rtemp, const float* __restrict__ diagb,
    const float* __restrict__ mstat, const float* __restrict__ lstat,
    float* __restrict__ o) {
  __shared__ float Pt[64][36];  // [t][s] transposed P tile (144B rows)
  __shared__ float Vs[32][68];  // [s][d] V chunk (272B rows)
  __shared__ float smax[32], srl[32];
  const int tid = threadIdx.x;
  const int wave = tid >> 5, lane = tid & 31;
  const int bh = blockIdx.y;
  const int b = bh / NH;
  const int h = bh % NH;
  const int tbase = blockIdx.x * 64;
  const int mrow = lane & 15;
  const int kk = (lane >> 4) * 2;
  const size_t base = ((size_t)b * TT * NH + h) * DH;
  const float scl = 0.125f * rtemp[bh];
  const float diag = diagb[bh];

  const int ssub1 = wave >> 2, tsub1 = wave & 3;  // phase-1 quadrant
  const int tsub2 = wave >> 1, dpair = wave & 1;  // phase-2 tile

  // Q fragments (B operand of score WMMA) are fixed for this block: preload.
  v2f qb[16];
  {
    const float* qrow = q + base + (size_t)(tbase + tsub1 * 16 + mrow) * HD;
#pragma unroll
    for (int ds = 0; ds < 16; ++ds) qb[ds] = *(const v2f*)&qrow[ds * 4 + kk];
  }

  v8f acc0 = {}, acc1 = {};
  for (int sc = 0; sc < TT; sc += 32) {
#pragma unroll
    for (int j = 0; j < 2; ++j) {  // stage 32x64 V chunk via async DMA
      int idx = tid + j * 256;
      int r = idx >> 4, c = (idx & 15) * 4;
      async_copy_b128(&Vs[r][c], &v[base + (size_t)(sc + r) * HD + c]);
    }
    if (tid < 32) {  // stage row stats
      smax[tid] = mstat[(size_t)bh * TT + sc + tid];
      srl[tid] = 1.0f / lstat[(size_t)bh * TT + sc + tid];
    }

    // phase 1: score tile (global K x register Q) while the V DMA flies
    v8f sacc = {};
    {
      const float* krow = k + base + (size_t)(sc + ssub1 * 16 + mrow) * HD;
#pragma unroll
      for (int ds = 0; ds < 16; ++ds) {
        v2f a = *(const v2f*)&krow[ds * 4 + kk];
        sacc = wmma_f32(a, qb[ds], sacc);
      }
    }
    wait_async();
    __syncthreads();
    {
      int t = tbase + tsub1 * 16 + mrow;
      bool mok = mask[b * TT + t] != 0;
#pragma unroll
      for (int i = 0; i < 8; ++i) {
        int sl = ssub1 * 16 + i + ((lane < 16) ? 0 : 8);
        int s = sc + sl;
        float val = sacc[i] * scl;
        if (s == t) val += diag;
        if (!mok) val = -1.0e9f;
        Pt[tsub1 * 16 + mrow][sl] = __expf(val - smax[sl]) * srl[sl];
      }
    }
    __syncthreads();

    // phase 2: O += P^T V, contraction over s in steps of 4
#pragma unroll
    for (int ks = 0; ks < 32; ks += 4) {
      v2f a = *(const v2f*)&Pt[tsub2 * 16 + mrow][ks + kk];
      v2f b0, b1;
      int dc0 = (dpair * 2) * 16 + mrow;
      int dc1 = (dpair * 2 + 1) * 16 + mrow;
      b0.x = Vs[ks + kk][dc0];
      b0.y = Vs[ks + kk + 1][dc0];
      b1.x = Vs[ks + kk][dc1];
      b1.y = Vs[ks + kk + 1][dc1];
      acc0 = wmma_f32(a, b0, acc0);
      acc1 = wmma_f32(a, b1, acc1);
    }
    __syncthreads();
  }
#pragma unroll
  for (int i = 0; i < 8; ++i) {
    int t = tbase + tsub2 * 16 + i + ((lane < 16) ? 0 : 8);
    int dc0 = (dpair * 2) * 16 + mrow;
    int dc1 = (dpair * 2 + 1) * 16 + mrow;
    o[base + (size_t)t * HD + dc0] = acc0[i];
    o[base + (size_t)t * HD + dc1] = acc1[i];
  }
}

extern "C" void kernel_launch(void* const* d_in, const int* in_sizes, int n_in,
                              void* d_out, int out_size, void* d_ws,
                              size_t ws_size, hipStream_t stream) {
  (void)in_sizes;
  (void)n_in;
  (void)out_size;
  (void)ws_size;
  const float* x = (const float*)d_in[0];
  const unsigned char* mask = (const unsigned char*)d_in[1];  // bool -> u8
  const float* p = (const float*)d_in[2];
  const float* wq = (const float*)d_in[3];
  const float* bq = (const float*)d_in[4];
  const float* wk = (const float*)d_in[5];
  const float* bk = (const float*)d_in[6];
  const float* wv = (const float*)d_in[7];
  const float* bv = (const float*)d_in[8];
  const float* wmod = (const float*)d_in[9];
  const float* bmod = (const float*)d_in[10];
  const float* wo = (const float*)d_in[11];
  const float* bo = (const float*)d_in[12];
  float* out = (float*)d_out;
  float* ws = (float*)d_ws;

  const size_t M = (size_t)NB * TT;  // 4096 rows
  float* q = ws;
  float* k = q + M * HD;
  float* v = k + M * HD;
  float* o = v + M * HD;
  float* mstat = o + M * HD;                     // [B*H][T]
  float* lstat = mstat + (size_t)NB * NH * TT;   // [B*H][T]
  float* rtemp = lstat + (size_t)NB * NH * TT;   // [B*H]
  float* diagb = rtemp + NB * NH;                // [B*H]

  dim3 gg(HD / 64, (int)(M / 64));  // (16, 64)
  gemm_bias_kernel<<<gg, 256, 0, stream>>>(x, wq, bq, q, (int)M, HD, HD);
  gemm_bias_kernel<<<gg, 256, 0, stream>>>(x, wk, bk, k, (int)M, HD, HD);
  gemm_bias_kernel<<<gg, 256, 0, stream>>>(x, wv, bv, v, (int)M, HD, HD);
  modulator_kernel<<<1, 32, 0, stream>>>(p, wmod, bmod, out + M * HD, rtemp,
                                         diagb);
  attn_stats_kernel<<<dim3(TT / 128, NB * NH), 256, 0, stream>>>(
      q, k, mask, rtemp, diagb, mstat, lstat);
  attn_out_kernel<<<dim3(TT / 64, NB * NH), 256, 0, stream>>>(
      q, k, v, mask, rtemp, diagb, mstat, lstat, o);
  gemm_bias_kernel<<<gg, 256, 0, stream>>>(o, wo, bo, out, (int)M, HD, HD);
}